// MTGNN_Layer_10677288698549
// MI455X (gfx1250) — compile-verified
//
#include <hip/hip_runtime.h>
#include <hip/hip_bf16.h>
#include <math.h>

// ---------------- problem constants (reference dims) ----------------
#define N_NODES 50000
#define D_FEAT  64
#define M_PATHS 3
#define E_EDGES 100000
#define L_PATH  4
#define H_HEADS 8
#define FH_HID  128
#define ALPHA_LRELU 0.01f
#define HD (H_HEADS * D_FEAT)   // 512

// ---------------- ws layout (in floats) ----------------
static constexpr long long OFF_FR   = 0;                                        // M*L*64 = 768 (pad to 1024)
static constexpr long long OFF_AMAX = 1024;                                     // M*N*H = 1,200,000
static constexpr long long OFF_DEN  = OFF_AMAX + (long long)M_PATHS*N_NODES*H_HEADS;
static constexpr long long OFF_A    = OFF_DEN  + (long long)M_PATHS*N_NODES*H_HEADS;  // M*E*H = 2,400,000
static constexpr long long OFF_ME   = OFF_A    + (long long)M_PATHS*E_EDGES*H_HEADS;  // M*E*D = 19,200,000
static constexpr long long OFF_Z    = OFF_ME   + (long long)M_PATHS*E_EDGES*D_FEAT;   // N*M*512 = 76,800,000
static constexpr long long OFF_WSUM = OFF_Z    + (long long)N_NODES*M_PATHS*HD;       // 4
static constexpr long long OFF_BETA = OFF_WSUM + 4;                                   // 4

// ---------------- WMMA types ----------------
typedef __attribute__((ext_vector_type(2))) float v2f;
typedef __attribute__((ext_vector_type(8))) float v8f;

// ---------------- helpers ----------------
__device__ inline void atomicMaxFloat(float* addr, float val) {
    // ordered-int trick: positive floats compare as ints, negative as reversed uints
    if (val >= 0.0f) atomicMax((int*)addr, __float_as_int(val));
    else             atomicMin((unsigned int*)addr, __float_as_uint(val));
}

// ---------------- fill ----------------
__global__ void k_fill(float* __restrict__ p, long long n, float v) {
    long long i = (long long)blockIdx.x * blockDim.x + threadIdx.x;
    if (i < n) p[i] = v;
}

// ---------------- rotation factors fr[m][l][dpair] ----------------
__global__ void k_fr(const float* __restrict__ r_vec, const int* __restrict__ etypes,
                     float* __restrict__ fr) {
    int m  = blockIdx.x;
    int dp = threadIdx.x;           // 0..31 complex pairs
    if (dp >= D_FEAT / 2) return;
    float cr = 1.0f, ci = 0.0f;     // identity at l = L-1
    fr[((long long)m * L_PATH + (L_PATH - 1)) * D_FEAT + 2 * dp]     = cr;
    fr[((long long)m * L_PATH + (L_PATH - 1)) * D_FEAT + 2 * dp + 1] = ci;
    for (int i = L_PATH - 2; i >= 0; --i) {
        int t = etypes[m * (L_PATH - 1) + i];
        int r = t >> 1;
        float re = r_vec[((long long)r * (D_FEAT / 2) + dp) * 2];
        float im = r_vec[((long long)r * (D_FEAT / 2) + dp) * 2 + 1];
        float inv = rsqrtf(re * re + im * im);
        re *= inv; im *= inv;
        if (t & 1) im = -im;        // conjugate entry of r_full
        float nr = cr * re - ci * im;
        float ni = cr * im + ci * re;
        cr = nr; ci = ni;
        fr[((long long)m * L_PATH + i) * D_FEAT + 2 * dp]     = cr;
        fr[((long long)m * L_PATH + i) * D_FEAT + 2 * dp + 1] = ci;
    }
}

// ---------------- pass 1: me, attention logits, segment max (wave per edge) ----------------
__global__ __launch_bounds__(256) void k_edge(const float* __restrict__ features,
                                              const float* __restrict__ attn1,
                                              const float* __restrict__ attn2,
                                              const float* __restrict__ fr,
                                              const int*   __restrict__ inst,
                                              float* __restrict__ me,
                                              float* __restrict__ a,
                                              float* __restrict__ amax) {
    __shared__ float s_fr[L_PATH * D_FEAT];      // 256
    __shared__ float s_a1[H_HEADS * D_FEAT];     // 512
    __shared__ float s_a2[H_HEADS * D_FEAT];     // 512
    int m   = blockIdx.y;
    int tid = threadIdx.x;
    for (int i = tid; i < L_PATH * D_FEAT; i += 256) s_fr[i] = fr[(long long)m * L_PATH * D_FEAT + i];
    for (int i = tid; i < H_HEADS * D_FEAT; i += 256) {
        s_a1[i] = attn1[(long long)m * H_HEADS * D_FEAT + i];
        s_a2[i] = attn2[(long long)m * H_HEADS * D_FEAT + i];
    }
    __syncthreads();

    long long e = (long long)blockIdx.x * 8 + (tid >> 5);
    if (e >= E_EDGES) return;
    int lane = tid & 31;
    int dp   = lane * 2;

    const int* ip = inst + ((long long)m * E_EDGES + e) * L_PATH;
    int nodes[L_PATH];
    #pragma unroll
    for (int l = 0; l < L_PATH; ++l) nodes[l] = ip[l];

    float accr = 0.0f, acci = 0.0f;
    #pragma unroll
    for (int l = 0; l < L_PATH; ++l) {
        float2 f = *(const float2*)(features + (long long)nodes[l] * D_FEAT + dp);
        float cr = s_fr[l * D_FEAT + dp];
        float ci = s_fr[l * D_FEAT + dp + 1];
        accr += f.x * cr - f.y * ci;
        acci += f.x * ci + f.y * cr;
    }
    accr *= (1.0f / L_PATH); acci *= (1.0f / L_PATH);
    *(float2*)(me + ((long long)m * E_EDGES + e) * D_FEAT + dp) = make_float2(accr, acci);

    int dst = nodes[0];
    float2 df = *(const float2*)(features + (long long)dst * D_FEAT + dp);
    #pragma unroll
    for (int h = 0; h < H_HEADS; ++h) {
        float p = df.x * s_a1[h * D_FEAT + dp] + df.y * s_a1[h * D_FEAT + dp + 1]
                + accr * s_a2[h * D_FEAT + dp] + acci * s_a2[h * D_FEAT + dp + 1];
        #pragma unroll
        for (int off = 16; off > 0; off >>= 1) p += __shfl_xor(p, off, 32);
        if (lane == h) {
            float av = p > 0.0f ? p : ALPHA_LRELU * p;
            a[((long long)m * E_EDGES + e) * H_HEADS + h] = av;
            atomicMaxFloat(&amax[((long long)m * N_NODES + dst) * H_HEADS + h], av);
        }
    }
}

// ---------------- pass 2: exp + segment sum ----------------
__global__ void k_den(const float* __restrict__ amax, float* __restrict__ a,
                      float* __restrict__ den, const int* __restrict__ inst) {
    long long i = (long long)blockIdx.x * blockDim.x + threadIdx.x;
    long long total = (long long)M_PATHS * E_EDGES * H_HEADS;
    if (i >= total) return;
    int h = (int)(i & (H_HEADS - 1));
    long long me_ = i >> 3;                       // m*E + e
    int m = (int)(me_ / E_EDGES);
    int dst = inst[me_ * L_PATH];
    float mx = amax[((long long)m * N_NODES + dst) * H_HEADS + h];
    if (!isfinite(mx)) mx = 0.0f;
    float ex = expf(a[i] - mx);
    a[i] = ex;                                     // overwrite logits with exp
    atomicAdd(&den[((long long)m * N_NODES + dst) * H_HEADS + h], ex);
}

// ---------------- pass 3: scatter attn * me into z (wave per edge) ----------------
__global__ __launch_bounds__(256) void k_agg(const float* __restrict__ me,
                                             const float* __restrict__ aex,
                                             const float* __restrict__ den,
                                             const int*   __restrict__ inst,
                                             float* __restrict__ z) {
    int m = blockIdx.y;
    long long e = (long long)blockIdx.x * 8 + (threadIdx.x >> 5);
    if (e >= E_EDGES) return;
    int lane = threadIdx.x & 31;
    int dp   = lane * 2;
    long long meo = (long long)m * E_EDGES + e;
    float2 mev = *(const float2*)(me + meo * D_FEAT + dp);
    int dst = inst[meo * L_PATH];
    float* zrow = z + ((long long)dst * M_PATHS + m) * HD;
    #pragma unroll
    for (int h = 0; h < H_HEADS; ++h) {
        float ex = aex[meo * H_HEADS + h];
        float dn = den[((long long)m * N_NODES + dst) * H_HEADS + h];
        float attn = ex / (dn > 0.0f ? dn : 1.0f);
        atomicAdd(zrow + h * D_FEAT + dp,     attn * mev.x);
        atomicAdd(zrow + h * D_FEAT + dp + 1, attn * mev.y);
    }
}

// ---------------- pass 4: elu in place ----------------
__global__ void k_elu(float* __restrict__ z, long long n) {
    long long i = (long long)blockIdx.x * blockDim.x + threadIdx.x;
    if (i >= n) return;
    float v = z[i];
    z[i] = v > 0.0f ? v : expf(v) - 1.0f;
}

// ---------------- WMMA GEMM: h = z @ W1^T + b1 ; w = tanh(h) @ W2 ; accumulate per-m mean
// One wave owns a DISTINCT 16-row tile of z (z read once from HBM) and all 8 col
// tiles of FH=128 via 8 accumulators; W1 fragments are L2-resident re-reads.
__global__ __launch_bounds__(256) void k_gemm(const float* __restrict__ z,
                                              const float* __restrict__ w1,
                                              const float* __restrict__ b1,
                                              const float* __restrict__ w2,
                                              float* __restrict__ wsum,
                                              int nrowtiles) {
    __shared__ float msum[M_PATHS];
    int tid = threadIdx.x;
    if (tid < M_PATHS) msum[tid] = 0.0f;
    __syncthreads();

    int wave = tid >> 5;
    int lane = tid & 31;
    int half = lane >> 4;                // K sub-pair select (rows v vs v+8 in C)
    int l    = lane & 15;
    int wtile = blockIdx.x * 8 + wave;   // global 16-row tile index over N*M rows
    bool active = (wtile < nrowtiles);

    if (active) {
        long long r0 = (long long)wtile * 16;
        const float* arow = z  + (r0 + l) * HD + 2 * half;   // A: rows r0..r0+15
        const float* bbase = w1 + (long long)l * HD + 2 * half;

        v8f c[8];
        #pragma unroll
        for (int ct = 0; ct < 8; ++ct) c[ct] = (v8f){};

        for (int k = 0; k < HD; k += 4) {
            v2f av = *(const v2f*)(arow + k);                // one A fragment...
            #pragma unroll
            for (int ct = 0; ct < 8; ++ct) {                 // ...feeds 8 WMMAs
                v2f bv = *(const v2f*)(bbase + (long long)ct * 16 * HD + k);
                c[ct] = __builtin_amdgcn_wmma_f32_16x16x4_f32(
                            false, av, false, bv, (short)0, c[ct], false, false);
            }
        }

        // epilogue: tanh(h + b1) * w2, reduce over 128 cols per row
        float p[8];
        #pragma unroll
        for (int v = 0; v < 8; ++v) p[v] = 0.0f;
        #pragma unroll
        for (int ct = 0; ct < 8; ++ct) {
            float bias = b1[ct * 16 + l];
            float wcol = w2[ct * 16 + l];
            #pragma unroll
            for (int v = 0; v < 8; ++v)
                p[v] += tanhf(c[ct][v] + bias) * wcol;
        }
        // butterfly within each 16-lane half (rows v in lanes 0-15, v+8 in 16-31)
        #pragma unroll
        for (int v = 0; v < 8; ++v) {
            #pragma unroll
            for (int off = 8; off > 0; off >>= 1)
                p[v] += __shfl_xor(p[v], off, 32);
        }
        if (l == 0) {                    // lane 0 -> rows 0..7, lane 16 -> rows 8..15
            #pragma unroll
            for (int v = 0; v < 8; ++v) {
                long long rr = r0 + v + 8 * half;    // flat row = n*M + m
                atomicAdd(&msum[(int)(rr % M_PATHS)], p[v]);
            }
        }
    }
    __syncthreads();
    if (tid < M_PATHS) atomicAdd(&wsum[tid], msum[tid]);
}

// ---------------- beta softmax ----------------
__global__ void k_beta(const float* __restrict__ wsum, float* __restrict__ beta) {
    if (threadIdx.x == 0 && blockIdx.x == 0) {
        float a0 = wsum[0] / (float)N_NODES;
        float a1 = wsum[1] / (float)N_NODES;
        float a2 = wsum[2] / (float)N_NODES;
        float mx = fmaxf(a0, fmaxf(a1, a2));
        float e0 = expf(a0 - mx), e1 = expf(a1 - mx), e2 = expf(a2 - mx);
        float s = e0 + e1 + e2;
        beta[0] = e0 / s; beta[1] = e1 / s; beta[2] = e2 / s;
    }
}

// ---------------- final: out = sum_m beta[m] * z[:, m, :] ----------------
__global__ void k_out(const float* __restrict__ z, const float* __restrict__ beta,
                      float* __restrict__ out) {
    long long i = (long long)blockIdx.x * blockDim.x + threadIdx.x;  // over N * 128 float4 groups
    long long total = (long long)N_NODES * (HD / 4);
    if (i >= total) return;
    long long n = i >> 7;
    int j = (int)(i & 127) * 4;
    float b0 = beta[0], b1v = beta[1], b2 = beta[2];
    const float* base = z + (long long)n * M_PATHS * HD + j;
    float4 v0 = *(const float4*)(base);
    float4 v1 = *(const float4*)(base + HD);
    float4 v2 = *(const float4*)(base + 2 * HD);
    float4 r;
    r.x = b0 * v0.x + b1v * v1.x + b2 * v2.x;
    r.y = b0 * v0.y + b1v * v1.y + b2 * v2.y;
    r.z = b0 * v0.z + b1v * v1.z + b2 * v2.z;
    r.w = b0 * v0.w + b1v * v1.w + b2 * v2.w;
    *(float4*)(out + n * HD + j) = r;
}

// ---------------- launcher ----------------
extern "C" void kernel_launch(void* const* d_in, const int* in_sizes, int n_in,
                              void* d_out, int out_size, void* d_ws, size_t ws_size,
                              hipStream_t stream) {
    const float* features = (const float*)d_in[0];
    const float* attn1_w  = (const float*)d_in[1];
    const float* attn2    = (const float*)d_in[2];
    const float* r_vec    = (const float*)d_in[3];
    const float* proj_w1  = (const float*)d_in[4];
    const float* proj_b1  = (const float*)d_in[5];
    const float* proj_w2  = (const float*)d_in[6];
    const int*   inst     = (const int*)d_in[7];
    const int*   etypes   = (const int*)d_in[8];
    float* out = (float*)d_out;
    float* ws  = (float*)d_ws;

    float* fr   = ws + OFF_FR;
    float* amax = ws + OFF_AMAX;
    float* den  = ws + OFF_DEN;
    float* a    = ws + OFF_A;
    float* me   = ws + OFF_ME;
    float* z    = ws + OFF_Z;
    float* wsum = ws + OFF_WSUM;
    float* beta = ws + OFF_BETA;

    const long long zsize   = (long long)N_NODES * M_PATHS * HD;   // 76.8M
    const long long nhsize  = (long long)M_PATHS * N_NODES * H_HEADS;
    const int TB = 256;
    auto blocks = [](long long n, int tb) { return (unsigned)((n + tb - 1) / tb); };

    // init (must run every call: atomics accumulate)
    k_fill<<<blocks(zsize, TB), TB, 0, stream>>>(z, zsize, 0.0f);
    k_fill<<<blocks(nhsize, TB), TB, 0, stream>>>(den, nhsize, 0.0f);
    k_fill<<<blocks(nhsize, TB), TB, 0, stream>>>(amax, nhsize, -INFINITY);
    k_fill<<<1, 32, 0, stream>>>(wsum, 8, 0.0f);

    // rotation factors
    k_fr<<<M_PATHS, 32, 0, stream>>>(r_vec, etypes, fr);

    // pass 1: me + logits + segment max
    dim3 g1((E_EDGES + 7) / 8, M_PATHS);
    k_edge<<<g1, TB, 0, stream>>>(features, attn1_w, attn2, fr, inst, me, a, amax);

    // pass 2: exp + segment sum
    long long dtot = (long long)M_PATHS * E_EDGES * H_HEADS;
    k_den<<<blocks(dtot, TB), TB, 0, stream>>>(amax, a, den, inst);

    // pass 3: scatter-aggregate into z
    k_agg<<<g1, TB, 0, stream>>>(me, a, den, inst, z);

    // pass 4: elu in place
    k_elu<<<blocks(zsize, TB), TB, 0, stream>>>(z, zsize);

    // WMMA projection + per-m score means: 150000/16 = 9375 row tiles, 8 waves/block
    int nrowtiles = (N_NODES * M_PATHS) / 16;
    k_gemm<<<(nrowtiles + 7) / 8, TB, 0, stream>>>(z, proj_w1, proj_b1, proj_w2, wsum, nrowtiles);

    // beta softmax
    k_beta<<<1, 32, 0, stream>>>(wsum, beta);

    // final combine
    long long otot = (long long)N_NODES * (HD / 4);
    k_out<<<blocks(otot, TB), TB, 0, stream>>>(z, beta, out);
}